// GraphSAGEModel_60086592471684
// MI455X (gfx1250) — compile-verified
//
#include <hip/hip_runtime.h>
#include <math.h>

typedef __attribute__((ext_vector_type(2))) float v2f;
typedef __attribute__((ext_vector_type(8))) float v8f;

#define F_IN  128
#define C_OUT 32

// ---------------------------------------------------------------------------
// Kernel 1: zero the scatter accumulators (acc[N*32] and deg[N], contiguous)
// ---------------------------------------------------------------------------
__global__ void sage_zero(float* __restrict__ p, long n) {
  long i = (long)blockIdx.x * blockDim.x + threadIdx.x;
  long stride = (long)gridDim.x * blockDim.x;
  for (; i < n; i += stride) p[i] = 0.0f;
}

// ---------------------------------------------------------------------------
// Kernel 2: fused dual GEMM via fp32 WMMA:
//   yl = x @ Wl^T   yr = x @ Wr^T     (x:[N,128], W:[32,128], y:[N,32])
// One wave handles a 16-row node tile and all 64 output columns
// (4 accumulator tiles). K marched in steps of 4 with V_WMMA_F32_16X16X4_F32.
//
// A 16x4 f32 layout : lane(0-15): {K=k,k+1} of row M=lane ; lane(16-31): {K=k+2,k+3}
// B 4x16  f32 layout: lane(0-15): {K=k,k+1} of col N=lane ; lane(16-31): {K=k+2,k+3}
// C/D 16x16 layout  : vgpr r -> (M=r, N=lo) lanes 0-15, (M=r+8, N=lo) lanes 16-31
// ---------------------------------------------------------------------------
__global__ void sage_gemm(const float* __restrict__ x,
                          const float* __restrict__ Wl,
                          const float* __restrict__ Wr,
                          float* __restrict__ yl,
                          float* __restrict__ yr,
                          int N) {
  const int wave = blockIdx.x * (blockDim.x >> 5) + (threadIdx.x >> 5);
  const int lane = threadIdx.x & 31;
  const int half = lane >> 4;      // 0: K pair {0,1}, 1: K pair {2,3}
  const int lo   = lane & 15;
  const long base = (long)wave * 16;
  if (base >= N) return;           // wave-uniform exit (EXEC stays all-1 for WMMA)

  long arow = base + lo;
  if (arow >= N) arow = N - 1;     // clamp loads for a partial tail tile
  const float* xrow = x + arow * F_IN;
  const float* wl0  = Wl + (long)lo        * F_IN;
  const float* wl1  = Wl + (long)(lo + 16) * F_IN;
  const float* wr0  = Wr + (long)lo        * F_IN;
  const float* wr1  = Wr + (long)(lo + 16) * F_IN;
  const int koff = 2 * half;

  v8f acc0 = {}, acc1 = {}, acc2 = {}, acc3 = {};
#pragma unroll 4
  for (int k = 0; k < F_IN; k += 4) {
    v2f a  = *(const v2f*)(xrow + k + koff);
    v2f b0 = *(const v2f*)(wl0  + k + koff);
    v2f b1 = *(const v2f*)(wl1  + k + koff);
    v2f b2 = *(const v2f*)(wr0  + k + koff);
    v2f b3 = *(const v2f*)(wr1  + k + koff);
    acc0 = __builtin_amdgcn_wmma_f32_16x16x4_f32(false, a, false, b0, (short)0, acc0, false, false);
    acc1 = __builtin_amdgcn_wmma_f32_16x16x4_f32(false, a, false, b1, (short)0, acc1, false, false);
    acc2 = __builtin_amdgcn_wmma_f32_16x16x4_f32(false, a, false, b2, (short)0, acc2, false, false);
    acc3 = __builtin_amdgcn_wmma_f32_16x16x4_f32(false, a, false, b3, (short)0, acc3, false, false);
  }

#pragma unroll
  for (int r = 0; r < 8; ++r) {
    long m = base + r + 8 * half;
    if (m < N) {
      yl[m * C_OUT + lo]      = acc0[r];
      yl[m * C_OUT + 16 + lo] = acc1[r];
      yr[m * C_OUT + lo]      = acc2[r];
      yr[m * C_OUT + 16 + lo] = acc3[r];
    }
  }
}

// ---------------------------------------------------------------------------
// Kernel 3: edge scatter. One wave per edge; lane c accumulates class c.
//   acc[dst, c] += yl[src, c];   deg[dst] += 1 (lane 0)
// All atomic targets (12.8 MB) are L2-resident.
// ---------------------------------------------------------------------------
__global__ void sage_scatter(const int* __restrict__ ei,   // [2, E]
                             const float* __restrict__ yl,
                             float* __restrict__ acc,
                             float* __restrict__ deg,
                             int E) {
  const int e    = blockIdx.x * (blockDim.x >> 5) + (threadIdx.x >> 5);
  const int lane = threadIdx.x & 31;
  if (e >= E) return;
  const long s = (long)ei[e];
  const long d = (long)ei[(long)E + e];
  const float v = yl[s * C_OUT + lane];
  atomicAdd(&acc[d * C_OUT + lane], v);
  if (lane == 0) atomicAdd(&deg[d], 1.0f);
}

// ---------------------------------------------------------------------------
// Kernel 4: per-node finalize. One wave per node, lane = class (C_OUT == 32
// == wave32 width). h = acc/max(deg,1) + b_l + yr ; relu ; log_softmax.
// ---------------------------------------------------------------------------
__global__ void sage_finalize(const float* __restrict__ acc,
                              const float* __restrict__ deg,
                              const float* __restrict__ yr,
                              const float* __restrict__ bl,
                              float* __restrict__ out,
                              int N) {
  const int n    = blockIdx.x * (blockDim.x >> 5) + (threadIdx.x >> 5);
  const int lane = threadIdx.x & 31;
  if (n >= N) return;
  float dg = deg[n];
  dg = dg > 1.0f ? dg : 1.0f;
  float h = acc[(long)n * C_OUT + lane] / dg + bl[lane] + yr[(long)n * C_OUT + lane];
  h = h > 0.0f ? h : 0.0f;

  // wave32 max-reduce
  float m = h;
#pragma unroll
  for (int off = 16; off > 0; off >>= 1)
    m = fmaxf(m, __shfl_xor(m, off, 32));
  // wave32 sum-reduce of exp(h - m)
  float ex = expf(h - m);
  float sum = ex;
#pragma unroll
  for (int off = 16; off > 0; off >>= 1)
    sum += __shfl_xor(sum, off, 32);

  out[(long)n * C_OUT + lane] = (h - m) - logf(sum);
}

// ---------------------------------------------------------------------------
// Launch
//   d_in: [0]=x (N*128 f32), [1]=edge_index (2*E int), [2]=W_l (32*128 f32),
//         [3]=b_l (32 f32), [4]=W_r (32*128 f32)
//   d_out: N*32 f32
//   d_ws:  yl[N*32] | yr[N*32] | acc[N*32] | deg[N]   (~38.8 MB)
// ---------------------------------------------------------------------------
extern "C" void kernel_launch(void* const* d_in, const int* in_sizes, int n_in,
                              void* d_out, int out_size, void* d_ws, size_t ws_size,
                              hipStream_t stream) {
  const float* x  = (const float*)d_in[0];
  const int*   ei = (const int*)d_in[1];
  const float* Wl = (const float*)d_in[2];
  const float* bl = (const float*)d_in[3];
  const float* Wr = (const float*)d_in[4];
  float* out = (float*)d_out;

  const int N = in_sizes[0] / F_IN;
  const int E = in_sizes[1] / 2;

  float* yl  = (float*)d_ws;
  float* yr  = yl  + (size_t)N * C_OUT;
  float* acc = yr  + (size_t)N * C_OUT;
  float* deg = acc + (size_t)N * C_OUT;

  // acc and deg are contiguous: zero N*32 + N floats in one pass
  const long zn = (long)N * C_OUT + N;
  sage_zero<<<4096, 256, 0, stream>>>(acc, zn);

  const int waves = (N + 15) / 16;                 // 16-row tiles
  sage_gemm<<<(waves + 1) / 2, 64, 0, stream>>>(x, Wl, Wr, yl, yr, N);

  sage_scatter<<<(E + 7) / 8, 256, 0, stream>>>(ei, yl, acc, deg, E);

  sage_finalize<<<(N + 7) / 8, 256, 0, stream>>>(acc, deg, yr, bl, out, N);
}